// DeltanetHead_20083267076224
// MI455X (gfx1250) — compile-verified
//
#include <hip/hip_runtime.h>
#include <hip/hip_bf16.h>

// ---------------------------------------------------------------------------
// DeltaNet-style head for MI455X (gfx1250, wave32, WMMA, async-to-LDS).
//   T=4096, B=8, C=1024, F=128
//   Phase 1: P[row, g*128+f] = x[row,:] . W_g[f,:] + b_g[f]   (6 gates fused)
//   Phase 2: sequential scan over T (per-batch workgroup, state in VGPRs)
//   Phase 3: y[row, c] = o[row,:] . O_w[c,:]
// FP32 matrix path via V_WMMA_F32_16X16X4_F32; global->LDS staging through
// GLOBAL_LOAD_ASYNC_TO_LDS_B128 with double buffering (ASYNCcnt pipelining).
// ---------------------------------------------------------------------------

#define T_DIM 4096
#define B_DIM 8
#define C_DIM 1024
#define F_DIM 128
#define TBROWS (T_DIM * B_DIM)      // 32768
#define NGATE 6
#define PCOLS (NGATE * F_DIM)       // 768

typedef float v2f __attribute__((ext_vector_type(2)));
typedef float v8f __attribute__((ext_vector_type(8)));
typedef int   v4i __attribute__((ext_vector_type(4)));

#define KC 32                        // K chunk staged through LDS
#define LDS_STRIDE (KC + 4)          // pad to dodge bank conflicts, keeps 16B align

// ---- gfx1250 async copy to LDS -------------------------------------------
#if defined(__has_builtin)
#if __has_builtin(__builtin_amdgcn_global_load_async_to_lds_b128)
#define HAVE_ASYNC_COPY 1
#endif
#endif
#ifndef HAVE_ASYNC_COPY
#define HAVE_ASYNC_COPY 0
#endif

#if HAVE_ASYNC_COPY
typedef __attribute__((address_space(1))) v4i* gv4i_ptr;
typedef __attribute__((address_space(3))) v4i* lv4i_ptr;
__device__ __forceinline__ void async_copy16(void* ldsDst, const void* gSrc) {
    // (global src, lds dst, imm offset, cpol) — emits global_load_async_to_lds_b128
    __builtin_amdgcn_global_load_async_to_lds_b128(
        (gv4i_ptr)gSrc, (lv4i_ptr)ldsDst, 0, 0);
}
#endif

template <int N>
__device__ __forceinline__ void wait_async() {
#if defined(__has_builtin)
#if __has_builtin(__builtin_amdgcn_s_wait_asynccnt)
    __builtin_amdgcn_s_wait_asynccnt(N);
#else
    asm volatile("s_wait_asynccnt %0" ::"i"(N) : "memory");
#endif
#else
    asm volatile("s_wait_asynccnt %0" ::"i"(N) : "memory");
#endif
}

struct ProjArgs {
    const float* x;                  // [TBROWS, C]
    const float* W[NGATE];           // each [F, C]
    const float* bias[NGATE];        // each [F]
    float* P;                        // [TBROWS, PCOLS]
};

// D = A*B + C, fp32, 16x16x4. A fragment: lane m = lane&15,
// v0/v1 = K{0,1} (lanes 0-15) or K{2,3} (lanes 16-31). B mirrored.
__device__ __forceinline__ v8f wmma_f32_k4(v2f a, v2f b, v8f c) {
    return __builtin_amdgcn_wmma_f32_16x16x4_f32(
        /*neg_a=*/false, a, /*neg_b=*/false, b,
        /*c_mod=*/(short)0, c, /*reuse_a=*/false, /*reuse_b=*/false);
}

// One KC-wide chunk of the 16x64 per-wave tile: 8 k-steps x 4 n-tiles.
__device__ __forceinline__ void gemm_chunk(const float* xsb, const float* wsb,
                                           int wave, int lane, v8f acc[4]) {
    const int mrow  = lane & 15;
    const int khalf = (lane >> 4) * 2;      // 0 or 2
    const float* xrow = &xsb[(wave * 16 + mrow) * LDS_STRIDE];
    #pragma unroll
    for (int ks = 0; ks < KC / 4; ++ks) {
        const int k0 = ks * 4 + khalf;
        float2 af = *(const float2*)&xrow[k0];
        v2f a; a.x = af.x; a.y = af.y;
        #pragma unroll
        for (int nt = 0; nt < 4; ++nt) {
            float2 bf = *(const float2*)&wsb[(nt * 16 + mrow) * LDS_STRIDE + k0];
            v2f b; b.x = bf.x; b.y = bf.y;
            acc[nt] = wmma_f32_k4(a, b, acc[nt]);
        }
    }
}

// ---------------------------------------------------------------------------
// Kernel 1: fused 6-gate projection GEMM.
// Block: 256 threads (8 waves). Block tile: 128 rows x 64 cols.
// grid = (TBROWS/128, PCOLS/64) = (256, 12); cols of one block share gate g.
// Double-buffered async global->LDS staging (6 async ops/thread/chunk).
// ---------------------------------------------------------------------------
__global__ __launch_bounds__(256) void proj_kernel(ProjArgs args) {
    __shared__ float xs[2][128 * LDS_STRIDE];   // x tile  [128 rows][KC]
    __shared__ float ws[2][64 * LDS_STRIDE];    // W tile  [64 cols][KC]

    const int tid  = threadIdx.x;
    const int wave = tid >> 5;
    const int lane = tid & 31;

    const int rowBase = blockIdx.x * 128;
    const int colBase = blockIdx.y * 64;
    const int g       = blockIdx.y >> 1;            // uniform gate per block
    const int fBase   = (blockIdx.y & 1) * 64;      // gate-local feature base
    const float* __restrict__ Wg = args.W[g];
    const float* __restrict__ bg = args.bias[g];

    // x tile: 256 thr * 4 float4 = 128 rows * 8 float4; W tile: 2 float4/thr
    const int xr = tid >> 3;            // base row 0..31 (stride 32 over 4 iters)
    const int xk = (tid & 7) * 4;       // k offset within chunk

    v8f acc[4] = {};

#if HAVE_ASYNC_COPY
    auto issue = [&](int buf, int kc) {
        const int kG = kc * KC;
        #pragma unroll
        for (int it = 0; it < 4; ++it) {
            const int r = xr + it * 32;
            async_copy16(&xs[buf][r * LDS_STRIDE + xk],
                         args.x + (size_t)(rowBase + r) * C_DIM + kG + xk);
        }
        #pragma unroll
        for (int it = 0; it < 2; ++it) {
            const int i   = tid + it * 256;
            const int col = i >> 3;
            const int kg  = (i & 7) * 4;
            async_copy16(&ws[buf][col * LDS_STRIDE + kg],
                         Wg + (size_t)(fBase + col) * C_DIM + kG + kg);
        }
    };
    issue(0, 0);
    for (int kc = 0; kc < C_DIM / KC; ++kc) {
        const int cur = kc & 1;
        if (kc + 1 < C_DIM / KC) {
            issue(cur ^ 1, kc + 1);     // prefetch next chunk into other buffer
            wait_async<6>();            // in-order: <=6 left => current 6 done
        } else {
            wait_async<0>();
        }
        __syncthreads();
        gemm_chunk(xs[cur], ws[cur], wave, lane, acc);
        __syncthreads();                // before next issue overwrites cur buf
    }
#else
    for (int kc = 0; kc < C_DIM / KC; ++kc) {
        const int kG = kc * KC;
        #pragma unroll
        for (int it = 0; it < 4; ++it) {
            const int r = xr + it * 32;
            const float4 v = *(const float4*)(args.x + (size_t)(rowBase + r) * C_DIM + kG + xk);
            *(float4*)&xs[0][r * LDS_STRIDE + xk] = v;
        }
        #pragma unroll
        for (int it = 0; it < 2; ++it) {
            const int i   = tid + it * 256;
            const int col = i >> 3;
            const int kg  = (i & 7) * 4;
            const float4 v = *(const float4*)(Wg + (size_t)(fBase + col) * C_DIM + kG + kg);
            *(float4*)&ws[0][col * LDS_STRIDE + kg] = v;
        }
        if (kc + 1 < C_DIM / KC)
            __builtin_prefetch(args.x + (size_t)(rowBase + xr) * C_DIM + kG + KC + xk, 0, 0);
        __syncthreads();
        gemm_chunk(xs[0], ws[0], wave, lane, acc);
        __syncthreads();
    }
#endif

    // Epilogue: bias + store. C/D layout: row = v + 8*(lane>=16), col = lane&15.
    const int n      = lane & 15;
    const int rowOff = (lane >> 4) * 8;
    #pragma unroll
    for (int nt = 0; nt < 4; ++nt) {
        const int colLocal = nt * 16 + n;
        const float bias = bg[fBase + colLocal];
        float* outp = args.P + (size_t)(rowBase + wave * 16 + rowOff) * PCOLS + colBase + colLocal;
        #pragma unroll
        for (int v = 0; v < 8; ++v)
            outp[(size_t)v * PCOLS] = acc[nt][v] + bias;
    }
}

// ---------------------------------------------------------------------------
// Kernel 2: sequential scan. One workgroup per batch (grid = 8), 512 threads.
// Thread owns state row r = tid>>2, columns [q*32, q*32+32) in registers.
//   st[r][j] <- st[r][j]*sig(i[j]) + (st[r]·a)*b[j]/sqrt(F) + c[r]*d[j]
//   out[j]    = sum_r sel[r]*st[r][j]
// Gate row (3 KB) for step t+1 is async-copied into the alternate LDS buffer
// while step t computes — hides L2 latency on the sequential critical path.
// Gate order in P row: 0=a 1=b 2=c 3=d 4=i 5=s(sel).
// ---------------------------------------------------------------------------
__global__ __launch_bounds__(512) void scan_kernel(const float* __restrict__ P,
                                                   const float* __restrict__ state_in,
                                                   float* __restrict__ o,
                                                   float* __restrict__ state_out) {
    const int bb   = blockIdx.x;
    const int tid  = threadIdx.x;
    const int r    = tid >> 2;          // state row 0..127
    const int q    = tid & 3;
    const int j0   = q * 32;            // column range base
    const int lane = tid & 31;
    const int wave = tid >> 5;          // 0..15

    __shared__ float gl[2][PCOLS];      // double-buffered gate vectors
    __shared__ float sigl[F_DIM];       // sigmoid(i)
    __shared__ float bsl[F_DIM];        // b / sqrt(F)
    __shared__ float red[16 * F_DIM];   // per-wave partial outs

    float st[32];
    {
        const float* sp = state_in + ((size_t)bb * F_DIM + r) * F_DIM + j0;
        #pragma unroll
        for (int jj = 0; jj < 32; jj += 4) {
            float4 v = *(const float4*)(sp + jj);
            st[jj] = v.x; st[jj+1] = v.y; st[jj+2] = v.z; st[jj+3] = v.w;
        }
    }
    const float inv_s = 0.08838834764831845f;   // 1/sqrt(128)

#if HAVE_ASYNC_COPY
    auto issueG = [&](int buf, int t) {
        const float* prow = P + ((size_t)t * B_DIM + bb) * PCOLS;
        if (tid < PCOLS / 4)            // 192 float4 slots
            async_copy16(&gl[buf][tid * 4], prow + tid * 4);
    };
    issueG(0, 0);
#endif

    for (int t = 0; t < T_DIM; ++t) {
        const int cur = t & 1;
#if HAVE_ASYNC_COPY
        if (t + 1 < T_DIM) {
            issueG(cur ^ 1, t + 1);     // prefetch next step's gates
            wait_async<1>();            // in-order: current step's copy done
        } else {
            wait_async<0>();
        }
#else
        {
            const float* prow = P + ((size_t)t * B_DIM + bb) * PCOLS;
            for (int idx = tid; idx < PCOLS; idx += 512)
                gl[cur][idx] = prow[idx];
        }
#endif
        __syncthreads();
        if (tid < F_DIM) {
            const float iv = gl[cur][4 * F_DIM + tid];
            sigl[tid] = 1.0f / (1.0f + __expf(-iv));
            bsl[tid]  = gl[cur][1 * F_DIM + tid] * inv_s;
        }
        __syncthreads();

        const float c_r   = gl[cur][2 * F_DIM + r];
        const float sel_r = gl[cur][5 * F_DIM + r];

        // v_r = st[r] · a   (4 lanes per row -> butterfly over xor 1,2)
        float v = 0.0f;
        #pragma unroll
        for (int jj = 0; jj < 32; ++jj) v += st[jj] * gl[cur][j0 + jj];
        v += __shfl_xor(v, 1);
        v += __shfl_xor(v, 2);

        // state update (registers) + per-column out partials
        #pragma unroll
        for (int jj = 0; jj < 32; ++jj) {
            const int j = j0 + jj;
            const float s_new = st[jj] * sigl[j] + v * bsl[j] + c_r * gl[cur][3 * F_DIM + j];
            st[jj] = s_new;
            float p = sel_r * s_new;            // reduce over 8 in-wave rows
            p += __shfl_xor(p, 4);
            p += __shfl_xor(p, 8);
            p += __shfl_xor(p, 16);
            if (lane < 4)                        // one lane per q holds wave sum
                red[wave * F_DIM + (lane & 3) * 32 + jj] = p;
        }
        __syncthreads();
        if (tid < F_DIM) {
            float s = 0.0f;
            #pragma unroll
            for (int w = 0; w < 16; ++w) s += red[w * F_DIM + tid];
            o[((size_t)t * B_DIM + bb) * F_DIM + tid] = s;
        }
        __syncthreads();
    }

    float* so = state_out + ((size_t)bb * F_DIM + r) * F_DIM + j0;
    #pragma unroll
    for (int jj = 0; jj < 32; jj += 4) {
        float4 v; v.x = st[jj]; v.y = st[jj+1]; v.z = st[jj+2]; v.w = st[jj+3];
        *(float4*)(so + jj) = v;
    }
}

// ---------------------------------------------------------------------------
// Kernel 3: y[row, c] = o[row,:] . O_w[c,:]   (K = 128, no bias)
// Same tiling as proj_kernel. grid = (TBROWS/128, C_DIM/64) = (256, 16).
// ---------------------------------------------------------------------------
__global__ __launch_bounds__(256) void out_gemm_kernel(const float* __restrict__ o,
                                                       const float* __restrict__ Ow,
                                                       float* __restrict__ y) {
    __shared__ float xs[2][128 * LDS_STRIDE];
    __shared__ float ws[2][64 * LDS_STRIDE];

    const int tid  = threadIdx.x;
    const int wave = tid >> 5;
    const int lane = tid & 31;
    const int rowBase = blockIdx.x * 128;
    const int colBase = blockIdx.y * 64;

    const int xr = tid >> 3;
    const int xk = (tid & 7) * 4;

    v8f acc[4] = {};

#if HAVE_ASYNC_COPY
    auto issue = [&](int buf, int kc) {
        const int kG = kc * KC;
        #pragma unroll
        for (int it = 0; it < 4; ++it) {
            const int r = xr + it * 32;
            async_copy16(&xs[buf][r * LDS_STRIDE + xk],
                         o + (size_t)(rowBase + r) * F_DIM + kG + xk);
        }
        #pragma unroll
        for (int it = 0; it < 2; ++it) {
            const int i   = tid + it * 256;
            const int col = i >> 3;
            const int kg  = (i & 7) * 4;
            async_copy16(&ws[buf][col * LDS_STRIDE + kg],
                         Ow + (size_t)(colBase + col) * F_DIM + kG + kg);
        }
    };
    issue(0, 0);
    for (int kc = 0; kc < F_DIM / KC; ++kc) {
        const int cur = kc & 1;
        if (kc + 1 < F_DIM / KC) {
            issue(cur ^ 1, kc + 1);
            wait_async<6>();
        } else {
            wait_async<0>();
        }
        __syncthreads();
        gemm_chunk(xs[cur], ws[cur], wave, lane, acc);
        __syncthreads();
    }
#else
    for (int kc = 0; kc < F_DIM / KC; ++kc) {
        const int kG = kc * KC;
        #pragma unroll
        for (int it = 0; it < 4; ++it) {
            const int r = xr + it * 32;
            const float4 v = *(const float4*)(o + (size_t)(rowBase + r) * F_DIM + kG + xk);
            *(float4*)&xs[0][r * LDS_STRIDE + xk] = v;
        }
        #pragma unroll
        for (int it = 0; it < 2; ++it) {
            const int i   = tid + it * 256;
            const int col = i >> 3;
            const int kg  = (i & 7) * 4;
            const float4 v = *(const float4*)(Ow + (size_t)(colBase + col) * F_DIM + kG + kg);
            *(float4*)&ws[0][col * LDS_STRIDE + kg] = v;
        }
        __syncthreads();
        gemm_chunk(xs[0], ws[0], wave, lane, acc);
        __syncthreads();
    }
#endif

    const int n      = lane & 15;
    const int rowOff = (lane >> 4) * 8;
    #pragma unroll
    for (int nt = 0; nt < 4; ++nt) {
        float* outp = y + (size_t)(rowBase + wave * 16 + rowOff) * C_DIM + colBase + nt * 16 + n;
        #pragma unroll
        for (int v = 0; v < 8; ++v)
            outp[(size_t)v * C_DIM] = acc[nt][v];
    }
}

// ---------------------------------------------------------------------------
extern "C" void kernel_launch(void* const* d_in, const int* in_sizes, int n_in,
                              void* d_out, int out_size, void* d_ws, size_t ws_size,
                              hipStream_t stream) {
    (void)in_sizes; (void)n_in; (void)out_size; (void)ws_size;

    const float* x     = (const float*)d_in[0];
    const float* state = (const float*)d_in[1];
    const float* Ow    = (const float*)d_in[14];

    ProjArgs pa;
    pa.x = x;
    // gate order: 0=A 1=B 2=C 3=D 4=I 5=S
    for (int g = 0; g < NGATE; ++g) {
        pa.W[g]    = (const float*)d_in[2 + 2 * g];
        pa.bias[g] = (const float*)d_in[3 + 2 * g];
    }

    float* P = (float*)d_ws;                                           // 32768*768*4   = 96 MiB
    float* o = (float*)((char*)d_ws + (size_t)TBROWS * PCOLS * 4);     // 32768*128*4   = 16 MiB
    pa.P = P;

    float* y         = (float*)d_out;                                  // [T,B,C]
    float* state_out = (float*)d_out + (size_t)TBROWS * C_DIM;         // [B,F,F]

    proj_kernel<<<dim3(TBROWS / 128, PCOLS / 64), 256, 0, stream>>>(pa);
    scan_kernel<<<dim3(B_DIM), 512, 0, stream>>>(P, state, o, state_out);
    out_gemm_kernel<<<dim3(TBROWS / 128, C_DIM / 64), 256, 0, stream>>>(o, Ow, y);
}